// FactorizationMachine_62981400429223
// MI455X (gfx1250) — compile-verified
//
#include <hip/hip_runtime.h>
#include <hip/hip_bf16.h>

typedef float v2f __attribute__((ext_vector_type(2)));
typedef float v8f __attribute__((ext_vector_type(8)));

#define K_DIM          128
#define ROWS_PER_BLOCK 16
#define LDS_PITCH      132   // 132 % 64 banks = 4 -> 16 rows hit distinct banks
#define BLOCK_THREADS  (ROWS_PER_BLOCK * 32)

__global__ __launch_bounds__(BLOCK_THREADS, 1)
void FactorizationMachine_62981400429223_kernel(
    const float* __restrict__ vals,
    const int*   __restrict__ row_idx,
    const int*   __restrict__ col_idx,
    const float* __restrict__ w0,
    const float* __restrict__ w,
    const float* __restrict__ v,
    float*       __restrict__ out,
    int nnz, int batch)
{
    __shared__ float xv_s[ROWS_PER_BLOCK][LDS_PITCH];
    __shared__ float lin_s[ROWS_PER_BLOCK];
    __shared__ float sq_s[ROWS_PER_BLOCK];
    __shared__ float diag_s[ROWS_PER_BLOCK];

    const int lane = threadIdx.x & 31;
    const int wave = threadIdx.x >> 5;          // 16 waves -> 16 batch rows
    const int b    = blockIdx.x * ROWS_PER_BLOCK + wave;

    // ---- locate this row's COO segment (row_idx is sorted) -----------------
    int start = 0, end = 0;
    {
        int lo = 0, hi = nnz;                   // lower_bound(b)
        while (lo < hi) { int mid = (lo + hi) >> 1;
                          if (row_idx[mid] <  b) lo = mid + 1; else hi = mid; }
        start = lo;
        hi = nnz;                               // lower_bound(b+1)
        while (lo < hi) { int mid = (lo + hi) >> 1;
                          if (row_idx[mid] <= b) lo = mid + 1; else hi = mid; }
        end = lo;
    }
    if (b >= batch) { start = end = 0; }

    // ---- gather/accumulate: lane owns k = 4*lane .. 4*lane+3 ----------------
    float ax0 = 0.f, ax1 = 0.f, ax2 = 0.f, ax3 = 0.f;   // xv partials
    float as0 = 0.f, as1 = 0.f, as2 = 0.f, as3 = 0.f;   // x@(v^2) partials
    float linp = 0.f;

    for (int base = start; base < end; base += 32) {
        const int j = base + lane;
        float vj = 0.f; int cj = 0;
        if (j < end) {
            vj = vals[j];
            cj = col_idx[j];
            linp += vj * w[cj];                 // linear term, lane-parallel
        }
        const int cnt = min(32, end - base);
        for (int i = 0; i < cnt; ++i) {
            const float val = __shfl(vj, i, 32);
            const int   col = __shfl(cj, i, 32);
            // coalesced 512B row read: 32 lanes x float4
            const float4 vv = *reinterpret_cast<const float4*>(
                v + (size_t)col * K_DIM + lane * 4);
            ax0 += val * vv.x;  as0 += val * vv.x * vv.x;
            ax1 += val * vv.y;  as1 += val * vv.y * vv.y;
            ax2 += val * vv.z;  as2 += val * vv.z * vv.z;
            ax3 += val * vv.w;  as3 += val * vv.w * vv.w;
        }
    }

    // ---- wave reductions for scalar terms ----------------------------------
    float sq = as0 + as1 + as2 + as3;
    for (int off = 16; off > 0; off >>= 1) {
        sq   += __shfl_xor(sq,   off, 32);
        linp += __shfl_xor(linp, off, 32);
    }

    // xv row -> LDS (16B-aligned: pitch 132*4 = 528 = 33*16)
    *reinterpret_cast<float4*>(&xv_s[wave][lane * 4]) =
        make_float4(ax0, ax1, ax2, ax3);
    if (lane == 0) { lin_s[wave] = linp; sq_s[wave] = sq; }

    __syncthreads();

    // ---- wave 0: Gram diagonal  diag[r] = <xv_r, xv_r>  via WMMA -----------
    if (wave == 0) {
        const int r    = lane & 15;             // M for A, N for B
        const int koff = (lane < 16) ? 0 : 2;   // K split across lane halves
        v8f gacc = {0.f, 0.f, 0.f, 0.f, 0.f, 0.f, 0.f, 0.f};
        #pragma unroll
        for (int kb = 0; kb < K_DIM; kb += 4) {
            v2f ab;
            ab[0] = xv_s[r][kb + koff];
            ab[1] = xv_s[r][kb + koff + 1];
            // D = A x B + C ; symmetric Gram: A and B lane layouts coincide
            gacc = __builtin_amdgcn_wmma_f32_16x16x4_f32(
                false, ab, false, ab, (short)0, gacc, false, false);
        }
        // diagonal (d,d): d<8 -> lane d, VGPR d ; d>=8 -> lane d+16, VGPR d-8
        #pragma unroll
        for (int i = 0; i < 8; ++i) {
            if (lane == i)      diag_s[i]     = gacc[i];
            if (lane == i + 24) diag_s[i + 8] = gacc[i];
        }
        // same wave reads back; compiler inserts s_wait_dscnt
        if (lane < ROWS_PER_BLOCK) {
            const int row = blockIdx.x * ROWS_PER_BLOCK + lane;
            if (row < batch) {
                out[row] = w0[0] + lin_s[lane]
                         + 0.5f * (diag_s[lane] - sq_s[lane]);
            }
        }
    }
}

extern "C" void kernel_launch(void* const* d_in, const int* in_sizes, int n_in,
                              void* d_out, int out_size, void* d_ws, size_t ws_size,
                              hipStream_t stream) {
    const float* vals    = (const float*)d_in[0];
    const int*   row_idx = (const int*)  d_in[1];
    const int*   col_idx = (const int*)  d_in[2];
    // d_in[3] = batch scalar on device; batch == out_size on host side
    const float* w0      = (const float*)d_in[4];
    const float* w       = (const float*)d_in[5];
    const float* v       = (const float*)d_in[6];
    float*       out     = (float*)d_out;

    const int nnz   = in_sizes[0];
    const int batch = out_size;
    const int grid  = (batch + ROWS_PER_BLOCK - 1) / ROWS_PER_BLOCK;

    FactorizationMachine_62981400429223_kernel
        <<<grid, BLOCK_THREADS, 0, stream>>>(
            vals, row_idx, col_idx, w0, w, v, out, nnz, batch);
}